// PCOSGraphNN_4071628997005
// MI455X (gfx1250) — compile-verified
//
#include <hip/hip_runtime.h>
#include <hip/hip_bf16.h>
#include <math.h>

// ---- problem constants (match reference) ----
#define NN    50000      // N_NODES
#define NE    400000     // N_EDGES
#define NG    64         // N_GRAPHS
#define FIN   128
#define HIDC  64         // HID
#define NHEAD 4
#define H4    256        // HEADS*HID

#define CDIV(a,b) (((a)+(b)-1)/(b))

// GEMM tiling
#define KC    128        // K-chunk staged in LDS
#define KPAD  132        // row stride in LDS: 132 % 64 == 4 -> conflict-free b64 reads

typedef float v2f __attribute__((ext_vector_type(2)));
typedef float v8f __attribute__((ext_vector_type(8)));

__device__ __forceinline__ float lrelu02(float x) { return x > 0.f ? x : 0.2f * x; }

// CAS-based float atomic max (handles negative values; init with -inf)
__device__ __forceinline__ void atomicMaxF(float* addr, float val) {
  unsigned* ua = (unsigned*)addr;
  unsigned cur = *ua;
  while (__uint_as_float(cur) < val) {
    unsigned prev = atomicCAS(ua, cur, __float_as_uint(val));
    if (prev == cur) break;
    cur = prev;
  }
}

// ---------------- utility ----------------
__global__ void k_fill(float* __restrict__ p, int n, float v) {
  int i = blockIdx.x * blockDim.x + threadIdx.x;
  if (i < n) p[i] = v;
}

// ---------------- edge MLP: ew = sigmoid(relu(ea@w1+b1)@w2+b2) ----------------
__global__ void k_edge_mlp(const float* __restrict__ ea, const float* __restrict__ w1,
                           const float* __restrict__ b1, const float* __restrict__ w2,
                           const float* __restrict__ b2, float* __restrict__ ew) {
  int e = blockIdx.x * blockDim.x + threadIdx.x;
  if (e >= NE) return;
  float a[8];
#pragma unroll
  for (int i = 0; i < 8; ++i) a[i] = ea[e * 8 + i];
  float o = b2[0];
#pragma unroll
  for (int j = 0; j < 8; ++j) {
    float hj = b1[j];
#pragma unroll
    for (int i = 0; i < 8; ++i) hj = fmaf(a[i], w1[i * 8 + j], hj);
    hj = fmaxf(hj, 0.f);
    o = fmaf(hj, w2[j], o);
  }
  ew[e] = 1.f / (1.f + expf(-o));
}

// ---------------- GCN degree / norm ----------------
__global__ void k_deg_scatter(const int* __restrict__ dst, const float* __restrict__ ew,
                              float* __restrict__ deg) {
  int e = blockIdx.x * blockDim.x + threadIdx.x;
  if (e >= NE) return;
  atomicAdd(&deg[dst[e]], ew[e]);
}

__global__ void k_dinv(const float* __restrict__ deg, float* __restrict__ dinv) {
  int i = blockIdx.x * blockDim.x + threadIdx.x;
  if (i >= NN) return;
  float d = deg[i];
  dinv[i] = d > 0.f ? rsqrtf(d) : 0.f;
}

// ---------------- fp32 WMMA GEMM: C[M,N] = A[M,K]@W[K,N] (+Cin) (+bias) (relu) ----
// Block = 256 threads = 8 waves. Block owns 8 consecutive 16-row M-tiles x one
// 64-col N slab. B panel staged in LDS in K-chunks of 128, transposed to
// Bs[col*KPAD + k] so each B fragment is ONE ds_load_b64 (rows k,k+1 of W).
// KPAD=132 (== 4 mod 64) makes the 32-lane b64 read pattern hit all 64 banks
// exactly once. Fragments for 4 k-steps are loaded into registers first, then
// 16 WMMAs issue back-to-back (one wait per 16 WMMAs instead of per WMMA).
// Wave-uniform control flow around WMMA: EXEC all-ones (ISA 7.12).
// Requires: M % 16 == 0, N % 64 == 0, K % 16 == 0.
__global__ void k_gemm_wmma(const float* __restrict__ A, const float* __restrict__ W,
                            const float* __restrict__ Cin, const float* __restrict__ bias,
                            float* __restrict__ C, int M, int N, int K, int relu) {
  __shared__ float Bs[64 * KPAD];          // 33,792 bytes

  int tilesM   = M >> 4;
  int bmBlocks = (tilesM + 7) >> 3;
  int bm = blockIdx.x % bmBlocks;          // which group of 8 M-tiles
  int tn = blockIdx.x / bmBlocks;          // which 64-col slab
  int waveId = threadIdx.x >> 5;
  int tm = bm * 8 + waveId;
  bool valid = (tm < tilesM);              // uniform per wave

  int lane = threadIdx.x & 31;
  int half = lane >> 4;                    // 0: K sub 0..1, 1: K sub 2..3
  int l16  = lane & 15;
  int kb   = half << 1;

  const float* Arow = A + (long long)(tm * 16 + l16) * K;

  v8f acc[4];
  if (valid && Cin) {
#pragma unroll
    for (int j = 0; j < 4; ++j)
#pragma unroll
      for (int v = 0; v < 8; ++v) {
        int row = tm * 16 + v + 8 * half;
        int col = tn * 64 + j * 16 + l16;
        acc[j][v] = Cin[(long long)row * N + col];
      }
  } else {
#pragma unroll
    for (int j = 0; j < 4; ++j)
#pragma unroll
      for (int v = 0; v < 8; ++v) acc[j][v] = 0.f;
  }

  for (int kc = 0; kc < K; kc += KC) {
    int kchunk = (K - kc) < KC ? (K - kc) : KC;

    __syncthreads();   // previous chunk's compute done before restaging
    // stage B chunk, transposed: Bs[c*KPAD + k] = W[kc+k][tn*64+c]
    for (int t = threadIdx.x; t < (kchunk << 6); t += 256) {
      int k = t >> 6, c = t & 63;
      Bs[c * KPAD + k] = W[(long long)(kc + k) * N + tn * 64 + c];
    }
    __syncthreads();

    if (valid) {
      int ksteps = kchunk >> 2;            // 16..32 WMMA k-steps in chunk
      for (int s0 = 0; s0 < ksteps; s0 += 4) {
        v2f af[4];
        v2f bf[4][4];
#pragma unroll
        for (int s = 0; s < 4; ++s)
          af[s] = *(const v2f*)(Arow + kc + ((s0 + s) << 2) + kb);
#pragma unroll
        for (int s = 0; s < 4; ++s)
#pragma unroll
          for (int j = 0; j < 4; ++j)
            bf[s][j] = *(const v2f*)&Bs[(j * 16 + l16) * KPAD + ((s0 + s) << 2) + kb];
#pragma unroll
        for (int s = 0; s < 4; ++s)
#pragma unroll
          for (int j = 0; j < 4; ++j)
            acc[j] = __builtin_amdgcn_wmma_f32_16x16x4_f32(
                false, af[s], false, bf[s][j], (short)0, acc[j], false, false);
      }
    }
  }

  if (valid) {
#pragma unroll
    for (int j = 0; j < 4; ++j)
#pragma unroll
      for (int v = 0; v < 8; ++v) {
        int row = tm * 16 + v + 8 * half;
        int col = tn * 64 + j * 16 + l16;
        float val = acc[j][v];
        if (bias) val += bias[col];
        if (relu) val = fmaxf(val, 0.f);
        C[(long long)row * N + col] = val;
      }
  }
}

// ---------------- GCN aggregation ----------------
__global__ void k_gcn_scatter(const int* __restrict__ src, const int* __restrict__ dst,
                              const float* __restrict__ ew, const float* __restrict__ dinv,
                              const float* __restrict__ hlin, float* __restrict__ hacc) {
  int t = blockIdx.x * blockDim.x + threadIdx.x;
  if (t >= NE * HIDC) return;
  int c = t & 63, e = t >> 6;
  int s = src[e], d = dst[e];
  float norm = dinv[s] * ew[e] * dinv[d];
  atomicAdd(&hacc[(long long)d * HIDC + c], norm * hlin[(long long)s * HIDC + c]);
}

__global__ void k_gcn_finalize(const float* __restrict__ dinv, const float* __restrict__ hlin,
                               const float* __restrict__ gcn_b, float* __restrict__ h) {
  int t = blockIdx.x * blockDim.x + threadIdx.x;
  if (t >= NN * HIDC) return;
  int c = t & 63, i = t >> 6;
  float di = dinv[i];
  float v = h[t] + di * di * hlin[t] + gcn_b[c];   // self-loop term + bias
  h[t] = fmaxf(v, 0.f);
}

// ---------------- GAT attention ----------------
__global__ void k_att(const float* __restrict__ hp, const float* __restrict__ att_s,
                      const float* __restrict__ att_d, float* __restrict__ as_,
                      float* __restrict__ ad_, float* __restrict__ mmax,
                      float* __restrict__ den) {
  int idx = blockIdx.x * blockDim.x + threadIdx.x;     // node*4 + head
  if (idx >= NN * NHEAD) return;
  int i = idx >> 2, h = idx & 3;
  const float* hrow = hp + (long long)i * H4 + h * HIDC;
  float s = 0.f, d = 0.f;
#pragma unroll 8
  for (int c = 0; c < HIDC; ++c) {
    s = fmaf(hrow[c], att_s[h * HIDC + c], s);
    d = fmaf(hrow[c], att_d[h * HIDC + c], d);
  }
  as_[idx] = s;
  ad_[idx] = d;
  mmax[idx] = -INFINITY;
  den[idx]  = 0.f;
}

// edges 0..NE-1 are real edges, NE..NE+NN-1 are self-loops
__global__ void k_gat_max(const int* __restrict__ src, const int* __restrict__ dst,
                          const float* __restrict__ as_, const float* __restrict__ ad_,
                          float* __restrict__ mmax) {
  int t = blockIdx.x * blockDim.x + threadIdx.x;
  if (t >= (NE + NN) * NHEAD) return;
  int h = t & 3, e = t >> 2;
  int s, d;
  if (e < NE) { s = src[e]; d = dst[e]; } else { s = d = e - NE; }
  float ee = lrelu02(as_[s * 4 + h] + ad_[d * 4 + h]);
  atomicMaxF(&mmax[d * 4 + h], ee);
}

__global__ void k_gat_den(const int* __restrict__ src, const int* __restrict__ dst,
                          const float* __restrict__ as_, const float* __restrict__ ad_,
                          const float* __restrict__ mmax, float* __restrict__ den) {
  int t = blockIdx.x * blockDim.x + threadIdx.x;
  if (t >= (NE + NN) * NHEAD) return;
  int h = t & 3, e = t >> 2;
  int s, d;
  if (e < NE) { s = src[e]; d = dst[e]; } else { s = d = e - NE; }
  float ee = lrelu02(as_[s * 4 + h] + ad_[d * 4 + h]);
  atomicAdd(&den[d * 4 + h], expf(ee - mmax[d * 4 + h]));
}

__global__ void k_gat_scatter(const int* __restrict__ src, const int* __restrict__ dst,
                              const float* __restrict__ as_, const float* __restrict__ ad_,
                              const float* __restrict__ mmax, const float* __restrict__ den,
                              const float* __restrict__ hp, float* __restrict__ h2) {
  int t = blockIdx.x * blockDim.x + threadIdx.x;
  if (t >= (NE + NN) * H4) return;
  int ch = t & 255, e = t >> 8;
  int h = ch >> 6;
  int s, d;
  if (e < NE) { s = src[e]; d = dst[e]; } else { s = d = e - NE; }
  float ee = lrelu02(as_[s * 4 + h] + ad_[d * 4 + h]);
  float alpha = expf(ee - mmax[d * 4 + h]) / den[d * 4 + h];
  atomicAdd(&h2[(long long)d * H4 + ch], alpha * hp[(long long)s * H4 + ch]);
}

__global__ void k_gat_finalize(const float* __restrict__ gat_b, float* __restrict__ h2) {
  int t = blockIdx.x * blockDim.x + threadIdx.x;
  if (t >= NN * H4) return;
  int ch = t & 255;
  h2[t] = fmaxf(h2[t] + gat_b[ch], 0.f);
}

// ---------------- GraphConv neighbor sum (original edges only) ----------------
__global__ void k_gc_scatter(const int* __restrict__ src, const int* __restrict__ dst,
                             const float* __restrict__ h2, float* __restrict__ agg) {
  int t = blockIdx.x * blockDim.x + threadIdx.x;
  if (t >= NE * H4) return;
  int ch = t & 255, e = t >> 8;
  atomicAdd(&agg[(long long)dst[e] * H4 + ch], h2[(long long)src[e] * H4 + ch]);
}

// ---------------- pooling ----------------
__global__ void k_pool(const float* __restrict__ h3, const int* __restrict__ batch,
                       float* __restrict__ cnt, float* __restrict__ psum,
                       float* __restrict__ pmax) {
  int t = blockIdx.x * blockDim.x + threadIdx.x;
  if (t >= NN * HIDC) return;
  int c = t & 63, i = t >> 6;
  int g = batch[i];
  float v = h3[t];
  atomicAdd(&psum[g * HIDC + c], v);
  atomicMaxF(&pmax[g * HIDC + c], v);
  if (c == 0) atomicAdd(&cnt[g], 1.f);
}

__global__ void k_zbuild(const float* __restrict__ cnt, const float* __restrict__ psum,
                         const float* __restrict__ pmax, float* __restrict__ z) {
  int t = blockIdx.x * blockDim.x + threadIdx.x;
  if (t >= NG * 2 * HIDC) return;
  int g = t >> 7, c = t & 127;
  z[t] = (c < HIDC) ? psum[g * HIDC + c] / fmaxf(cnt[g], 1.f)
                    : pmax[g * HIDC + (c - HIDC)];
}

// ---------------- small FC: Y[rows,kout] = relu?(X[rows,kin]@W[kin,kout]+b) ----
__global__ void k_fc(const float* __restrict__ X, const float* __restrict__ W,
                     const float* __restrict__ b, float* __restrict__ Y,
                     int rows, int kin, int kout, int relu) {
  int t = blockIdx.x * blockDim.x + threadIdx.x;
  if (t >= rows * kout) return;
  int g = t / kout, j = t % kout;
  float s = b[j];
  for (int k = 0; k < kin; ++k) s = fmaf(X[g * kin + k], W[k * kout + j], s);
  if (relu) s = fmaxf(s, 0.f);
  Y[t] = s;
}

// ================================================================
extern "C" void kernel_launch(void* const* d_in, const int* in_sizes, int n_in,
                              void* d_out, int out_size, void* d_ws, size_t ws_size,
                              hipStream_t stream) {
  const float* x         = (const float*)d_in[0];
  const float* edge_attr = (const float*)d_in[1];
  const int*   eidx      = (const int*)d_in[2];
  const int*   batch     = (const int*)d_in[3];
  const float* gcn_w     = (const float*)d_in[4];
  const float* gcn_b     = (const float*)d_in[5];
  const float* gat_w     = (const float*)d_in[6];
  const float* att_s     = (const float*)d_in[7];
  const float* att_d     = (const float*)d_in[8];
  const float* gat_b     = (const float*)d_in[9];
  const float* gc_w_rel  = (const float*)d_in[10];
  const float* gc_w_root = (const float*)d_in[11];
  const float* gc_b      = (const float*)d_in[12];
  const float* em_w1     = (const float*)d_in[13];
  const float* em_b1     = (const float*)d_in[14];
  const float* em_w2     = (const float*)d_in[15];
  const float* em_b2     = (const float*)d_in[16];
  const float* fc1_w     = (const float*)d_in[17];
  const float* fc1_b     = (const float*)d_in[18];
  const float* fc2_w     = (const float*)d_in[19];
  const float* fc2_b     = (const float*)d_in[20];
  const float* fc3_w     = (const float*)d_in[21];
  const float* fc3_b     = (const float*)d_in[22];
  float* out = (float*)d_out;

  const int* src = eidx;
  const int* dst = eidx + NE;

  // ---- workspace carve-out (~133 MB, aliased) ----
  float* ws = (float*)d_ws;
  size_t off = 0;
  float* ew   = ws + off; off += NE;
  float* deg  = ws + off; off += NN;
  float* dinv = ws + off; off += NN;
  float* as_  = ws + off; off += (size_t)NN * NHEAD;
  float* ad_  = ws + off; off += (size_t)NN * NHEAD;
  float* mmax = ws + off; off += (size_t)NN * NHEAD;
  float* den  = ws + off; off += (size_t)NN * NHEAD;
  float* cnt  = ws + off; off += NG;
  float* psum = ws + off; off += NG * HIDC;
  float* pmax = ws + off; off += NG * HIDC;
  float* z    = ws + off; off += NG * 2 * HIDC;
  float* z1   = ws + off; off += NG * HIDC;
  float* z2   = ws + off; off += NG * 32;
  float* bufA = ws + off; off += (size_t)NN * HIDC;  // hlin, later h3tmp
  float* bufB = ws + off; off += (size_t)NN * HIDC;  // h,    later h3
  float* bufC = ws + off; off += (size_t)NN * H4;    // hp,   later agg
  float* bufD = ws + off; off += (size_t)NN * H4;    // h2

  const int T = 256;
  auto gemm = [&](const float* A, const float* W, const float* Cin, const float* bias,
                  float* C, int M, int Nn, int K, int relu) {
    int tilesM = M / 16;
    int bmBlocks = CDIV(tilesM, 8);
    int blocks = bmBlocks * (Nn / 64);
    k_gemm_wmma<<<blocks, 256, 0, stream>>>(A, W, Cin, bias, C, M, Nn, K, relu);
  };

  // 1. edge MLP
  k_edge_mlp<<<CDIV(NE, T), T, 0, stream>>>(edge_attr, em_w1, em_b1, em_w2, em_b2, ew);

  // 2. GCN degree (self-loop weight 1) + sym norm
  k_fill<<<CDIV(NN, T), T, 0, stream>>>(deg, NN, 1.0f);
  k_deg_scatter<<<CDIV(NE, T), T, 0, stream>>>(dst, ew, deg);
  k_dinv<<<CDIV(NN, T), T, 0, stream>>>(deg, dinv);

  // 3. hlin = x @ gcn_w   [50000,128]x[128,64]
  gemm(x, gcn_w, nullptr, nullptr, bufA, NN, HIDC, FIN, 0);

  // 4. GCN weighted aggregation + self loop + bias + relu
  k_fill<<<CDIV(NN * HIDC, T), T, 0, stream>>>(bufB, NN * HIDC, 0.f);
  k_gcn_scatter<<<CDIV(NE * HIDC, T), T, 0, stream>>>(src, dst, ew, dinv, bufA, bufB);
  k_gcn_finalize<<<CDIV(NN * HIDC, T), T, 0, stream>>>(dinv, bufA, gcn_b, bufB);

  // 5. hp = h @ gat_w   [50000,64]x[64,256]
  gemm(bufB, gat_w, nullptr, nullptr, bufC, NN, H4, HIDC, 0);

  // 6. GAT attention (3-pass segment softmax + scatter)
  k_att<<<CDIV(NN * NHEAD, T), T, 0, stream>>>(bufC, att_s, att_d, as_, ad_, mmax, den);
  k_gat_max<<<CDIV((NE + NN) * NHEAD, T), T, 0, stream>>>(src, dst, as_, ad_, mmax);
  k_gat_den<<<CDIV((NE + NN) * NHEAD, T), T, 0, stream>>>(src, dst, as_, ad_, mmax, den);
  k_fill<<<CDIV(NN * H4, T), T, 0, stream>>>(bufD, NN * H4, 0.f);
  k_gat_scatter<<<CDIV((NE + NN) * H4, T), T, 0, stream>>>(src, dst, as_, ad_, mmax, den,
                                                           bufC, bufD);
  k_gat_finalize<<<CDIV(NN * H4, T), T, 0, stream>>>(gat_b, bufD);

  // 7. GraphConv: agg = sum_in h2 ; h3 = relu(agg@W_rel + h2@W_root + b)
  k_fill<<<CDIV(NN * H4, T), T, 0, stream>>>(bufC, NN * H4, 0.f);   // reuse hp as agg
  k_gc_scatter<<<CDIV(NE * H4, T), T, 0, stream>>>(src, dst, bufD, bufC);
  gemm(bufC, gc_w_rel, nullptr, nullptr, bufA, NN, HIDC, H4, 0);           // h3tmp
  gemm(bufD, gc_w_root, bufA, gc_b, bufB, NN, HIDC, H4, 1);                // h3 (fused)

  // 8. global mean+max pooling
  k_fill<<<1, T, 0, stream>>>(cnt, NG, 0.f);
  k_fill<<<CDIV(NG * HIDC, T), T, 0, stream>>>(psum, NG * HIDC, 0.f);
  k_fill<<<CDIV(NG * HIDC, T), T, 0, stream>>>(pmax, NG * HIDC, -INFINITY);
  k_pool<<<CDIV(NN * HIDC, T), T, 0, stream>>>(bufB, batch, cnt, psum, pmax);
  k_zbuild<<<CDIV(NG * 2 * HIDC, T), T, 0, stream>>>(cnt, psum, pmax, z);

  // 9. head MLP
  k_fc<<<CDIV(NG * HIDC, T), T, 0, stream>>>(z, fc1_w, fc1_b, z1, NG, 2 * HIDC, HIDC, 1);
  k_fc<<<CDIV(NG * 32, T), T, 0, stream>>>(z1, fc2_w, fc2_b, z2, NG, HIDC, 32, 1);
  k_fc<<<1, T, 0, stream>>>(z2, fc3_w, fc3_b, out, NG, 32, 2, 0);
}